// CustomGNN_42039139893788
// MI455X (gfx1250) — compile-verified
//
#include <hip/hip_runtime.h>
#include <hip/hip_bf16.h>
#include <math.h>

typedef float v2f __attribute__((ext_vector_type(2)));
typedef float v8f __attribute__((ext_vector_type(8)));

#define HEADS 4
#define HC 256          // HEADS * 64 (both layers: H*Hid and H*Out)
#define CDIM 64
#define NEG_SLOPE 0.2f

// ---------------------------------------------------------------------------
// FP32 WMMA GEMM: C[M x Ncol] = A[M x K] * B[K x Ncol]
// Each wave computes a 16x64 strip of C (4 accumulator tiles) so the A
// fragment is loaded once per K-step and reused by 4 back-to-back
// V_WMMA_F32_16X16X4_F32 issues (independent accumulators -> no RAW stalls).
// Requires M % 16 == 0, Ncol % 64 == 0, K % 8 == 0 (holds: 50000/128/64/256).
// Fragment layouts per CDNA5 ISA 7.12.2:
//   A 16x4 (2 VGPRs): lanes 0-15 hold K={k,k+1}, lanes 16-31 K={k+2,k+3},
//                     M = lane & 15.
//   B 4x16 (2 VGPRs): N = lane & 15; VGPR0 -> K = k + 2*laneHi,
//                     VGPR1 -> K = k+1 + 2*laneHi.
//   C/D 16x16 (8 VGPRs): acc[i] is row (i + 8*laneHi), col (lane & 15).
// ---------------------------------------------------------------------------
__global__ void gemm_wmma_f32(const float* __restrict__ A,
                              const float* __restrict__ B,
                              float* __restrict__ C,
                              int M, int K, int Ncol) {
    const int lane   = threadIdx.x & 31;
    const int wave   = threadIdx.x >> 5;
    const int wtile  = blockIdx.x * 4 + wave;     // one 16x64 strip per wave
    const int wtprow = Ncol >> 6;                 // wave strips per row of C
    const int total  = (M >> 4) * wtprow;
    if (wtile >= total) return;                   // whole-wave exit; EXEC full
    const int trow   = wtile / wtprow;
    const int tcol   = wtile - trow * wtprow;
    const int row0   = trow << 4;
    const int col0   = tcol << 6;
    const int laneM  = lane & 15;
    const int laneHi = lane >> 4;                 // 0 or 1

    const float* Arow  = A + (size_t)(row0 + laneM) * K;
    const float* Bbase = B + (size_t)col0 + laneM;

    v8f acc0 = {}, acc1 = {}, acc2 = {}, acc3 = {};
#pragma unroll 2
    for (int k = 0; k < K; k += 4) {
        const int ka = k + (laneHi << 1);
        v2f a;
        a.x = Arow[ka];
        a.y = Arow[ka + 1];
        const float* B0 = Bbase + (size_t)ka * Ncol;        // K = ka
        const float* B1 = Bbase + (size_t)(ka + 1) * Ncol;  // K = ka+1
        v2f b0, b1, b2, b3;
        b0.x = B0[0];  b0.y = B1[0];
        b1.x = B0[16]; b1.y = B1[16];
        b2.x = B0[32]; b2.y = B1[32];
        b3.x = B0[48]; b3.y = B1[48];
        acc0 = __builtin_amdgcn_wmma_f32_16x16x4_f32(false, a, false, b0, (short)0, acc0, false, false);
        acc1 = __builtin_amdgcn_wmma_f32_16x16x4_f32(false, a, false, b1, (short)0, acc1, false, false);
        acc2 = __builtin_amdgcn_wmma_f32_16x16x4_f32(false, a, false, b2, (short)0, acc2, false, false);
        acc3 = __builtin_amdgcn_wmma_f32_16x16x4_f32(false, a, false, b3, (short)0, acc3, false, false);
    }

    float* Crow = C + (size_t)(row0 + 8 * laneHi) * Ncol + col0 + laneM;
#pragma unroll
    for (int i = 0; i < 8; ++i) {
        float* r = Crow + (size_t)i * Ncol;
        r[0]  = acc0[i];
        r[16] = acc1[i];
        r[32] = acc2[i];
        r[48] = acc3[i];
    }
}

// ---------------------------------------------------------------------------
// Per-(node, head) attention scores:  as[n,h] = <h[n,h,:], att_src[h,:]>
// ---------------------------------------------------------------------------
__global__ void attn_scores(const float* __restrict__ Hm,
                            const float* __restrict__ att_s,
                            const float* __restrict__ att_d,
                            float* __restrict__ as_, float* __restrict__ ad_,
                            int N) {
    const int idx = blockIdx.x * blockDim.x + threadIdx.x;
    if (idx >= N * HEADS) return;
    const int n = idx >> 2, h = idx & 3;
    const float* row = Hm + (size_t)n * HC + h * CDIM;
    const float* vs  = att_s + h * CDIM;
    const float* vd  = att_d + h * CDIM;
    float s = 0.f, d = 0.f;
#pragma unroll 8
    for (int c = 0; c < CDIM; ++c) {
        const float v = row[c];
        s = fmaf(v, vs[c], s);
        d = fmaf(v, vd[c], d);
    }
    as_[idx] = s;
    ad_[idx] = d;
}

// Order-preserving float <-> uint map so segment-max can use u32 atomics.
__device__ __forceinline__ unsigned fmap(float x) {
    unsigned u = __float_as_uint(x);
    return (u & 0x80000000u) ? ~u : (u | 0x80000000u);
}
__device__ __forceinline__ float funmap(unsigned u) {
    u = (u & 0x80000000u) ? (u ^ 0x80000000u) : ~u;
    return __uint_as_float(u);
}

__device__ __forceinline__ void edge_nodes(int eid, int E,
                                           const int* __restrict__ srcv,
                                           const int* __restrict__ dstv,
                                           int& s, int& d) {
    if (eid < E) { s = srcv[eid]; d = dstv[eid]; }
    else         { s = d = eid - E; }            // appended self loops
}

__device__ __forceinline__ float leaky(float e) {
    return e > 0.f ? e : NEG_SLOPE * e;
}

// ---------------------------------------------------------------------------
// Pass 1: segment max of LeakyReLU(a_src[src]+a_dst[dst]) over dst
// ---------------------------------------------------------------------------
__global__ void edge_max(const int* __restrict__ srcv, const int* __restrict__ dstv,
                         const float* __restrict__ as_, const float* __restrict__ ad_,
                         unsigned* __restrict__ dmax, int E, int N) {
    const int idx = blockIdx.x * blockDim.x + threadIdx.x;
    const int tot = (E + N) * HEADS;
    if (idx >= tot) return;
    const int eid = idx >> 2, h = idx & 3;
    int s, d;
    edge_nodes(eid, E, srcv, dstv, s, d);
    const float e = leaky(as_[s * HEADS + h] + ad_[d * HEADS + h]);
    atomicMax(&dmax[d * HEADS + h], fmap(e));
}

// ---------------------------------------------------------------------------
// Pass 2: ex = exp(e - max[dst]); cache ex per edge; segment sum over dst
// ---------------------------------------------------------------------------
__global__ void edge_expsum(const int* __restrict__ srcv, const int* __restrict__ dstv,
                            const float* __restrict__ as_, const float* __restrict__ ad_,
                            const unsigned* __restrict__ dmax,
                            float* __restrict__ EXv, float* __restrict__ dsum,
                            int E, int N) {
    const int idx = blockIdx.x * blockDim.x + threadIdx.x;
    const int tot = (E + N) * HEADS;
    if (idx >= tot) return;
    const int eid = idx >> 2, h = idx & 3;
    int s, d;
    edge_nodes(eid, E, srcv, dstv, s, d);
    const float e  = leaky(as_[s * HEADS + h] + ad_[d * HEADS + h]);
    const float m  = funmap(dmax[d * HEADS + h]);
    const float ex = __expf(e - m);
    EXv[idx] = ex;
    atomicAdd(&dsum[d * HEADS + h], ex);
}

// ---------------------------------------------------------------------------
// Pass 3: acc[dst,h,c] += h[src,h,c] * (ex / (sum[dst,h]+1e-16))
// One 256-thread block per edge: coalesced 1KB gather + 1KB atomic scatter.
// ---------------------------------------------------------------------------
__global__ void edge_scatter(const int* __restrict__ srcv, const int* __restrict__ dstv,
                             const float* __restrict__ Hm,
                             const float* __restrict__ EXv,
                             const float* __restrict__ dsum,
                             float* __restrict__ ACC, int E, int N) {
    const int eid = blockIdx.x;
    const int t   = threadIdx.x;                 // t = h*64 + c
    int s, d;
    edge_nodes(eid, E, srcv, dstv, s, d);
    const int h = t >> 6;
    const float alpha = EXv[eid * HEADS + h] / (dsum[d * HEADS + h] + 1e-16f);
    atomicAdd(&ACC[(size_t)d * HC + t], Hm[(size_t)s * HC + t] * alpha);
}

// ---------------------------------------------------------------------------
// Head mean + bias (+ optional ReLU)
// ---------------------------------------------------------------------------
__global__ void head_mean(const float* __restrict__ ACC,
                          const float* __restrict__ bias,
                          float* __restrict__ out, int N, int do_relu) {
    const int idx = blockIdx.x * blockDim.x + threadIdx.x;
    if (idx >= N * CDIM) return;
    const int n = idx >> 6, c = idx & 63;
    const float* r = ACC + (size_t)n * HC;
    float v = (r[c] + r[CDIM + c] + r[2 * CDIM + c] + r[3 * CDIM + c]) * 0.25f
              + bias[c];
    if (do_relu) v = fmaxf(v, 0.f);
    out[idx] = v;
}

// ---------------------------------------------------------------------------
extern "C" void kernel_launch(void* const* d_in, const int* in_sizes, int n_in,
                              void* d_out, int out_size, void* d_ws, size_t ws_size,
                              hipStream_t stream) {
    const float* x        = (const float*)d_in[0];
    const int*   ei       = (const int*)  d_in[1];
    const float* W1       = (const float*)d_in[2];
    const float* att_src1 = (const float*)d_in[3];
    const float* att_dst1 = (const float*)d_in[4];
    const float* b1       = (const float*)d_in[5];
    const float* W2       = (const float*)d_in[6];
    const float* att_src2 = (const float*)d_in[7];
    const float* att_dst2 = (const float*)d_in[8];
    const float* b2       = (const float*)d_in[9];

    const int Fin = 128;
    const int N   = in_sizes[0] / Fin;           // 50000
    const int E   = in_sizes[1] / 2;             // 800000
    const int ET  = E + N;                       // edges + self loops
    const int* srcv = ei;
    const int* dstv = ei + E;

    // Workspace layout (floats)
    float* ws    = (float*)d_ws;
    float* Hbuf  = ws;                            // [N, 256]
    float* ACC   = Hbuf + (size_t)N * HC;         // [N, 256]
    float* HID   = ACC  + (size_t)N * HC;         // [N, 64]
    float* ASRC  = HID  + (size_t)N * CDIM;       // [N, 4]
    float* ADST  = ASRC + (size_t)N * HEADS;      // [N, 4]
    unsigned* DMAX = (unsigned*)(ADST + (size_t)N * HEADS);   // [N, 4]
    float* DSUM  = (float*)(DMAX + (size_t)N * HEADS);        // [N, 4]
    float* EXv   = DSUM + (size_t)N * HEADS;      // [ET, 4]

    const int gemmStrips = (N / 16) * (HC / 64);  // 16x64 strips (one per wave)
    const int gemmBlocks = (gemmStrips + 3) / 4;  // 4 waves per 128-thread block
    const int nhBlocks   = (N * HEADS + 255) / 256;
    const int ehBlocks   = (ET * HEADS + 255) / 256;
    const int ncBlocks   = (N * CDIM + 255) / 256;

    // ---------------- Layer 1 ----------------
    hipMemsetAsync(DMAX, 0, (size_t)N * HEADS * 4, stream);  // 0 == fmap floor
    hipMemsetAsync(DSUM, 0, (size_t)N * HEADS * 4, stream);
    hipMemsetAsync(ACC,  0, (size_t)N * HC * 4, stream);

    gemm_wmma_f32<<<gemmBlocks, 128, 0, stream>>>(x, W1, Hbuf, N, Fin, HC);
    attn_scores<<<nhBlocks, 256, 0, stream>>>(Hbuf, att_src1, att_dst1, ASRC, ADST, N);
    edge_max<<<ehBlocks, 256, 0, stream>>>(srcv, dstv, ASRC, ADST, DMAX, E, N);
    edge_expsum<<<ehBlocks, 256, 0, stream>>>(srcv, dstv, ASRC, ADST, DMAX, EXv, DSUM, E, N);
    edge_scatter<<<ET, HC, 0, stream>>>(srcv, dstv, Hbuf, EXv, DSUM, ACC, E, N);
    head_mean<<<ncBlocks, 256, 0, stream>>>(ACC, b1, HID, N, /*relu=*/1);

    // ---------------- Layer 2 ----------------
    hipMemsetAsync(DMAX, 0, (size_t)N * HEADS * 4, stream);
    hipMemsetAsync(DSUM, 0, (size_t)N * HEADS * 4, stream);
    hipMemsetAsync(ACC,  0, (size_t)N * HC * 4, stream);

    gemm_wmma_f32<<<gemmBlocks, 128, 0, stream>>>(HID, W2, Hbuf, N, CDIM, HC);
    attn_scores<<<nhBlocks, 256, 0, stream>>>(Hbuf, att_src2, att_dst2, ASRC, ADST, N);
    edge_max<<<ehBlocks, 256, 0, stream>>>(srcv, dstv, ASRC, ADST, DMAX, E, N);
    edge_expsum<<<ehBlocks, 256, 0, stream>>>(srcv, dstv, ASRC, ADST, DMAX, EXv, DSUM, E, N);
    edge_scatter<<<ET, HC, 0, stream>>>(srcv, dstv, Hbuf, EXv, DSUM, ACC, E, N);
    head_mean<<<ncBlocks, 256, 0, stream>>>(ACC, b2, (float*)d_out, N, /*relu=*/0);
}